// _RoisPairExpandingLayer_6399501271857
// MI455X (gfx1250) — compile-verified
//
#include <hip/hip_runtime.h>
#include <hip/hip_bf16.h>
#include <stdint.h>

// Problem constants (static shapes per the reference).
#define BIMG 16
#define KOBJ 768
#define PAIRS ((KOBJ * (KOBJ - 1)) / 2)   // 294528
#define RPI   (KOBJ + PAIRS)              // rows per image = 295296
#define K5    (KOBJ * 5)                  // dwords of one image's rois = 3840
#define QUADS (PAIRS / 4)                 // 73632 (PAIRS divisible by 4)

typedef __attribute__((ext_vector_type(4))) float v4f;

// Types matching the async builtin's parameters: pointer to 16B int vector,
// in global (AS1) / LDS (AS3) address spaces.
typedef int v4i __attribute__((vector_size(16)));
typedef __attribute__((address_space(1))) v4i gv4i;   // global int4
typedef __attribute__((address_space(3))) v4i lv4i;   // LDS int4

__device__ __forceinline__ int tri_rowstart(int i) {
    // number of (i<j) pairs in rows < i, row-major triu order
    return (i * (2 * KOBJ - 1 - i)) >> 1;
}

// Union rows (the hot kernel; placed first so its asm shows in the snippet).
// One image per blockIdx.y; stage that image's 15 KB of rois into LDS via the
// gfx1250 async-to-LDS path, then each thread emits 4 consecutive pairs
// = 80 B via five 16B-aligned non-temporal b128 stores.
__global__ __launch_bounds__(256) void union_pairs_kernel(
    const float* __restrict__ rois, float* __restrict__ out) {
    __shared__ alignas(16) float lds[K5];
    const int b = blockIdx.y;
    const float* src = rois + (size_t)b * K5;

#if __has_builtin(__builtin_amdgcn_global_load_async_to_lds_b128)
    // Async DMA-style copy: 960 x b128, ASYNCcnt-tracked, no data VGPRs.
    for (int t = threadIdx.x; t < K5 / 4; t += 256) {
        __builtin_amdgcn_global_load_async_to_lds_b128(
            (gv4i*)(src + 4 * t), (lv4i*)(&lds[4 * t]), 0, 0);
    }
#if __has_builtin(__builtin_amdgcn_s_wait_asynccnt)
    __builtin_amdgcn_s_wait_asynccnt(0);
#else
    asm volatile("s_wait_asynccnt 0" ::: "memory");
#endif
#else
    for (int t = threadIdx.x; t < K5; t += 256) lds[t] = src[t];
#endif
    __syncthreads();

    const int quad = blockIdx.x * 256 + threadIdx.x;
    if (quad >= QUADS) return;
    const int p0 = quad * 4;

    // Decode row i: largest i with tri_rowstart(i) <= p0.
    // disc = (2K-1)^2 - 8*p0; at row starts disc = (2K-1-2i)^2 exactly
    // (integer < 2^22, exact in f32), so sqrtf is exact there; fix-up loops
    // absorb any interior rounding.
    const float A = (float)(2 * KOBJ - 1);
    float disc = (float)((2 * KOBJ - 1) * (2 * KOBJ - 1) - 8 * p0);
    int i = (int)((A - __builtin_sqrtf(disc)) * 0.5f);
    if (i < 0) i = 0;
    if (i > KOBJ - 2) i = KOBJ - 2;
    while (i > 0 && tri_rowstart(i) > p0) --i;
    while (i < KOBJ - 2 && tri_rowstart(i + 1) <= p0) ++i;
    int j = i + 1 + (p0 - tri_rowstart(i));

    const float fb = (float)b;
    float v[20];
#pragma unroll
    for (int q = 0; q < 4; ++q) {
        const float* ri = &lds[i * 5];
        const float* rj = &lds[j * 5];
        v[q * 5 + 0] = fb;
        v[q * 5 + 1] = fminf(ri[1], rj[1]);   // min x1
        v[q * 5 + 2] = fminf(ri[2], rj[2]);   // min y1
        v[q * 5 + 3] = fmaxf(ri[3], rj[3]);   // max x2
        v[q * 5 + 4] = fmaxf(ri[4], rj[4]);   // max y2
        if (++j == KOBJ) { ++i; j = i + 1; }  // roll into next triu row
    }

    // 4 pairs * 20 B = 80 B; base is 16B-aligned (RPI*20, K*20, p0*20 all %16==0).
    float* o = out + ((size_t)b * RPI + KOBJ + (size_t)p0) * 5;
    v4f* o4 = (v4f*)o;
#pragma unroll
    for (int t = 0; t < 5; ++t) {
        v4f val = {v[4 * t + 0], v[4 * t + 1], v[4 * t + 2], v[4 * t + 3]};
        __builtin_nontemporal_store(val, o4 + t);
    }
}

// The first K rows of each image block are a verbatim copy of the input rois
// rows (input col0 already equals the image index as float).
__global__ __launch_bounds__(256) void coords_copy_kernel(
    const float* __restrict__ rois, float* __restrict__ out) {
    int d = blockIdx.x * 256 + threadIdx.x;           // dword id, 0..B*K*5-1
    if (d >= BIMG * K5) return;
    int row = d / 5;
    int c   = d - row * 5;
    int b   = row / KOBJ;
    int k   = row - b * KOBJ;
    float v = rois[d];
    __builtin_nontemporal_store(v, out + ((size_t)b * RPI + k) * 5 + c);
}

extern "C" void kernel_launch(void* const* d_in, const int* in_sizes, int n_in,
                              void* d_out, int out_size, void* d_ws, size_t ws_size,
                              hipStream_t stream) {
    const float* rois = (const float*)d_in[0];
    float* out = (float*)d_out;

    // Union rows: 73632 quads per image, 16 images.
    dim3 grid((QUADS + 255) / 256, BIMG);
    union_pairs_kernel<<<grid, dim3(256), 0, stream>>>(rois, out);

    // Copy B*K rois rows into their per-image slots.
    coords_copy_kernel<<<dim3((BIMG * K5 + 255) / 256), dim3(256), 0, stream>>>(rois, out);
}